// GSCAN_model_22978075034370
// MI455X (gfx1250) — compile-verified
//
#include <hip/hip_runtime.h>

typedef __attribute__((ext_vector_type(2))) float v2f;
typedef __attribute__((ext_vector_type(4))) float v4f;
typedef __attribute__((ext_vector_type(8))) float v8f;

#define C_IN 17
#define EMB 64
#define WAVES_PER_BLOCK 8
#define CELLS_PER_WAVE 16
#define CELLS_PER_BLOCK (WAVES_PER_BLOCK * CELLS_PER_WAVE)  // 128

// One wave32 computes a 16-cell x 64-feature output tile with five
// V_WMMA_F32_16X16X4_F32 ops (fp32 end-to-end; no precision downgrade).
__global__ __launch_bounds__(256) void gscan_embed_kernel(
    const float* __restrict__ situation,   // [ncells, 17]
    const float* __restrict__ Wsize,       // [4, 16]
    const float* __restrict__ Wshape,      // [4, 16]
    const float* __restrict__ Wrgb,        // [4, 16]
    const float* __restrict__ Wagent,      // [5, 16]
    float* __restrict__ out,               // [ncells, 64]
    long long ncells)
{
    __shared__ __align__(16) float lds_tile[WAVES_PER_BLOCK][CELLS_PER_WAVE * EMB]; // 32 KB
    __shared__ float lds_mask[WAVES_PER_BLOCK][CELLS_PER_WAVE];

    const int lane  = threadIdx.x & 31;
    const int wave  = threadIdx.x >> 5;
    const long long tile     = (long long)blockIdx.x * WAVES_PER_BLOCK + wave;
    const long long cellbase = tile * CELLS_PER_WAVE;
    if (cellbase >= ncells) return;   // grid sized exactly; defensive only

    const int m     = lane & 15;      // A-fragment row (cell within tile)
    const int n     = lane & 15;      // B/C-fragment column
    const int khalf = lane >> 4;      // 0: K=0,1   1: K=2,3
    const int k0    = khalf * 2;

    const float* row = situation + (cellbase + m) * C_IN;

    // ---- A fragments (16x4 fp32 each; agent split into two K=4 steps) ----
    v2f a_size, a_shape, a_rgb, a_ag0, a_ag1;
    a_size.x  = row[0  + k0]; a_size.y  = row[0  + k0 + 1];
    a_shape.x = row[4  + k0]; a_shape.y = row[4  + k0 + 1];
    a_rgb.x   = row[8  + k0]; a_rgb.y   = row[8  + k0 + 1];
    a_ag0.x   = row[12 + k0]; a_ag0.y   = row[12 + k0 + 1];
    // agent-hi step covers logical K=4..7 -> channels 16,17,18,19 (only 16 real)
    a_ag1.x   = (khalf == 0) ? row[16] : 0.0f;
    a_ag1.y   = 0.0f;

    // ---- B fragments (4x16 fp32, N across lanes, K split by lane-half) ----
    v2f b_size, b_shape, b_rgb, b_ag0, b_ag1;
    b_size.x  = Wsize [k0 * 16 + n];  b_size.y  = Wsize [(k0 + 1) * 16 + n];
    b_shape.x = Wshape[k0 * 16 + n];  b_shape.y = Wshape[(k0 + 1) * 16 + n];
    b_rgb.x   = Wrgb  [k0 * 16 + n];  b_rgb.y   = Wrgb  [(k0 + 1) * 16 + n];
    b_ag0.x   = Wagent[k0 * 16 + n];  b_ag0.y   = Wagent[(k0 + 1) * 16 + n];
    b_ag1.x   = (khalf == 0) ? Wagent[4 * 16 + n] : 0.0f;  // K=4 row; K=5..7 zero
    b_ag1.y   = 0.0f;

    // ---- Stage fallback (raw channels then zero-pad) into LDS tile ----
    float* tileptr = lds_tile[wave];
    {
        v4f z = {0.0f, 0.0f, 0.0f, 0.0f};
        v4f* t4 = (v4f*)tileptr;
        #pragma unroll
        for (int r = 0; r < 8; ++r) t4[r * 32 + lane] = z;   // zero 16x64 tile
    }
    // Each of the 17 channels is held by exactly one lane (per cell): write it.
    tileptr[m * EMB + 0  + k0]     = a_size.x;
    tileptr[m * EMB + 0  + k0 + 1] = a_size.y;
    tileptr[m * EMB + 4  + k0]     = a_shape.x;
    tileptr[m * EMB + 4  + k0 + 1] = a_shape.y;
    tileptr[m * EMB + 8  + k0]     = a_rgb.x;
    tileptr[m * EMB + 8  + k0 + 1] = a_rgb.y;
    tileptr[m * EMB + 12 + k0]     = a_ag0.x;
    tileptr[m * EMB + 12 + k0 + 1] = a_ag0.y;
    if (khalf == 0) tileptr[m * EMB + 16] = a_ag1.x;

    // ---- Channel-sum mask: per-lane partial + cross-half shuffle ----
    float partial = a_size.x + a_size.y + a_shape.x + a_shape.y +
                    a_rgb.x + a_rgb.y + a_ag0.x + a_ag0.y + a_ag1.x;
    float full = partial + __shfl_xor(partial, 16, 32);
    lds_mask[wave][m] = full;   // lanes m and m+16 write identical value

    // ---- Five fp32 WMMAs: D = A x B (+C) ----
    v8f c0 = {}, c1 = {}, c2 = {}, c3 = {};
    c0 = __builtin_amdgcn_wmma_f32_16x16x4_f32(false, a_size,  false, b_size,
                                               (short)0, c0, false, false);
    c1 = __builtin_amdgcn_wmma_f32_16x16x4_f32(false, a_shape, false, b_shape,
                                               (short)0, c1, false, false);
    c2 = __builtin_amdgcn_wmma_f32_16x16x4_f32(false, a_rgb,   false, b_rgb,
                                               (short)0, c2, false, false);
    c3 = __builtin_amdgcn_wmma_f32_16x16x4_f32(false, a_ag0,   false, b_ag0,
                                               (short)0, c3, false, false);
    c3 = __builtin_amdgcn_wmma_f32_16x16x4_f32(false, a_ag1,   false, b_ag1,
                                               (short)0, c3, false, false);

    // ---- Overwrite embedded features where mask>0 (wave-private LDS) ----
    #pragma unroll
    for (int r = 0; r < 8; ++r) {
        const int mrow = r + 8 * khalf;          // C/D layout: VGPR r -> M=r(+8)
        if (lds_mask[wave][mrow] > 0.0f) {
            tileptr[mrow * EMB +  0 + n] = c0[r];
            tileptr[mrow * EMB + 16 + n] = c1[r];
            tileptr[mrow * EMB + 32 + n] = c2[r];
            tileptr[mrow * EMB + 48 + n] = c3[r];
        }
    }

    // ---- Coalesced b128 writeback of the finished tile ----
    v4f* o4 = (v4f*)(out + cellbase * EMB);
    v4f* t4 = (v4f*)tileptr;
    #pragma unroll
    for (int r = 0; r < 8; ++r) {
        o4[r * 32 + lane] = t4[r * 32 + lane];
    }
}

extern "C" void kernel_launch(void* const* d_in, const int* in_sizes, int n_in,
                              void* d_out, int out_size, void* d_ws, size_t ws_size,
                              hipStream_t stream) {
    const float* situation = (const float*)d_in[0];
    const float* Wsize     = (const float*)d_in[1];
    const float* Wshape    = (const float*)d_in[2];
    const float* Wrgb      = (const float*)d_in[3];
    const float* Wagent    = (const float*)d_in[4];
    float* out = (float*)d_out;

    const long long ncells = (long long)in_sizes[0] / C_IN;   // 2048*32*32
    const int blocks = (int)((ncells + CELLS_PER_BLOCK - 1) / CELLS_PER_BLOCK);

    gscan_embed_kernel<<<blocks, 256, 0, stream>>>(
        situation, Wsize, Wshape, Wrgb, Wagent, out, ncells);
}